// MoELayerStacks_22677427322892
// MI455X (gfx1250) — compile-verified
//
#include <hip/hip_runtime.h>
#include <hip/hip_bf16.h>

#define L2N 15
#define L3N 32
#define NUM_EXPERTS 16
#define EXPERT_DIM 2048
#define ROUTER_DIM 128
#define BTOK 32768
#define TILES_PER_EXPERT (BTOK / 16)

typedef __attribute__((ext_vector_type(2))) float v2f;
typedef __attribute__((ext_vector_type(8))) float v8f;

__device__ __forceinline__ float clamp01(float v) {
    return fminf(fmaxf(v, 0.0f), 1.0f);
}

// ---------------------------------------------------------------------------
// Kernel 0: zero routing state (counts). cursor/offsets written by scan.
// ---------------------------------------------------------------------------
__global__ __launch_bounds__(64) void zero_kernel(int* counts) {
    if (threadIdx.x < NUM_EXPERTS) counts[threadIdx.x] = 0;
}

// ---------------------------------------------------------------------------
// Kernel 1: router logits + argmax (first-max tiebreak, matches jnp.argmax),
//           plus per-expert histogram.
// ---------------------------------------------------------------------------
__global__ __launch_bounds__(256) void router_kernel(
    const float* __restrict__ rin, const float* __restrict__ rw,
    const float* __restrict__ rb, int* __restrict__ eidx,
    int* __restrict__ counts)
{
    __shared__ float ws[NUM_EXPERTS][ROUTER_DIM];
    __shared__ float bs[NUM_EXPERTS];
    for (int i = threadIdx.x; i < NUM_EXPERTS * ROUTER_DIM; i += 256)
        ws[i / ROUTER_DIM][i % ROUTER_DIM] = rw[i];
    if (threadIdx.x < NUM_EXPERTS) bs[threadIdx.x] = rb[threadIdx.x];
    __syncthreads();

    const int t = blockIdx.x * 256 + threadIdx.x;
    const float* __restrict__ xr = rin + (size_t)t * ROUTER_DIM;

    float acc[NUM_EXPERTS];
#pragma unroll
    for (int e = 0; e < NUM_EXPERTS; ++e) acc[e] = bs[e];

    for (int k = 0; k < ROUTER_DIM; k += 4) {
        const float4 x4 = *(const float4*)(xr + k);
#pragma unroll
        for (int e = 0; e < NUM_EXPERTS; ++e) {
            acc[e] += x4.x * ws[e][k + 0];
            acc[e] += x4.y * ws[e][k + 1];
            acc[e] += x4.z * ws[e][k + 2];
            acc[e] += x4.w * ws[e][k + 3];
        }
    }
    int best = 0;
    float bv = acc[0];
#pragma unroll
    for (int e = 1; e < NUM_EXPERTS; ++e)
        if (acc[e] > bv) { bv = acc[e]; best = e; }
    eidx[t] = best;
    atomicAdd(&counts[best], 1);
}

// ---------------------------------------------------------------------------
// Kernel 2: 16-entry exclusive scan + cursor init (single thread; trivial).
// ---------------------------------------------------------------------------
__global__ __launch_bounds__(32) void scan_kernel(const int* __restrict__ counts,
                                                  int* __restrict__ offsets,
                                                  int* __restrict__ cursor) {
    if (threadIdx.x == 0) {
        int run = 0;
        for (int e = 0; e < NUM_EXPERTS; ++e) {
            offsets[e] = run;
            cursor[e] = run;
            run += counts[e];
        }
        offsets[NUM_EXPERTS] = run;
    }
}

// ---------------------------------------------------------------------------
// Kernel 3: counting-sort scatter: token ids grouped by expert.
// ---------------------------------------------------------------------------
__global__ __launch_bounds__(256) void scatter_kernel(const int* __restrict__ eidx,
                                                      int* __restrict__ cursor,
                                                      int* __restrict__ perm) {
    const int t = blockIdx.x * 256 + threadIdx.x;
    const int e = eidx[t];
    const int pos = atomicAdd(&cursor[e], 1);
    perm[pos] = t;
}

// ---------------------------------------------------------------------------
// Kernel 4: fused expert MLP. One wave32 per 16-token tile of one expert.
//
// l1 (2048 -> 16) done as 512x V_WMMA_F32_16X16X4_F32:
//   A tile 16x4 fp32: lane (m = lane&15, hi = lane>>4) holds
//     A[M=m][K=2*hi], A[M=m][K=2*hi+1]  -> one b64 load from token m's row.
//   B tile 4x16 fp32: lane holds B[K=2*hi..2*hi+1][N=m] = l1_w[e][m][k+2*hi..]
//     -> one b64 load (l1_w inner dim contiguous).
// C/D 16x16 fp32: VGPR r, lane -> D[M=r+8*hi][N=m]; M = token-in-tile, N = out.
// ---------------------------------------------------------------------------
__global__ __launch_bounds__(32) void moe_mlp_kernel(
    const float* __restrict__ x,
    const float* __restrict__ l1_w, const float* __restrict__ l1_b,
    const float* __restrict__ l2_w, const float* __restrict__ l2_b,
    const float* __restrict__ out_w, const float* __restrict__ out_b,
    const int* __restrict__ offsets, const int* __restrict__ perm,
    float* __restrict__ out)
{
    const int e   = blockIdx.y;
    const int beg = offsets[e];
    const int end = offsets[e + 1];
    const int row0 = beg + blockIdx.x * 16;
    if (row0 >= end) return;                 // uniform exit: EXEC all-1s for WMMA

    const int lane = threadIdx.x;
    const int m  = lane & 15;
    const int hi = lane >> 4;

    __shared__ float l1s[16][17];            // [token][neuron] (padded)
    __shared__ float red[16][16];            // [token][m] l3 partials

    const int rowm = row0 + m;
    const int tok  = (rowm < end) ? perm[rowm] : perm[row0];  // safe dup gather

    const float* __restrict__ arow = x    + (size_t)tok * EXPERT_DIM + 2 * hi;
    const float* __restrict__ brow = l1_w + ((size_t)(e * 16 + m)) * EXPERT_DIM + 2 * hi;

    v8f acc0 = {};
    v8f acc1 = {};
    for (int k = 0; k < EXPERT_DIM; k += 16) {
        const v2f a0 = *(const v2f*)(arow + k + 0);
        const v2f b0 = *(const v2f*)(brow + k + 0);
        const v2f a1 = *(const v2f*)(arow + k + 4);
        const v2f b1 = *(const v2f*)(brow + k + 4);
        const v2f a2 = *(const v2f*)(arow + k + 8);
        const v2f b2 = *(const v2f*)(brow + k + 8);
        const v2f a3 = *(const v2f*)(arow + k + 12);
        const v2f b3 = *(const v2f*)(brow + k + 12);
        acc0 = __builtin_amdgcn_wmma_f32_16x16x4_f32(false, a0, false, b0, (short)0, acc0, false, false);
        acc1 = __builtin_amdgcn_wmma_f32_16x16x4_f32(false, a1, false, b1, (short)0, acc1, false, false);
        acc0 = __builtin_amdgcn_wmma_f32_16x16x4_f32(false, a2, false, b2, (short)0, acc0, false, false);
        acc1 = __builtin_amdgcn_wmma_f32_16x16x4_f32(false, a3, false, b3, (short)0, acc1, false, false);
    }

    // l1c = acc + bias -> LDS [token][neuron]
    const float bias1 = l1_b[e * 16 + m];
#pragma unroll
    for (int r = 0; r < 8; ++r)
        l1s[r + 8 * hi][m] = acc0[r] + acc1[r] + bias1;
    __syncthreads();   // single-wave WG: lowers to cheap sync, orders LDS

    // l2: K padded 30->32, N=32 as two 16-wide WMMA tiles.
    const float* __restrict__ w2 = l2_w + (size_t)e * L3N * (2 * L2N);
    v8f c0 = {};
    v8f c1 = {};
    for (int k = 0; k < 32; k += 4) {
        const int ka = k + 2 * hi;
        v2f a, bA, bB;
        // activation: [0,15): clip(v^2 * 255/256), [15,30): clip(v), [30,32): 0
#pragma unroll
        for (int j = 0; j < 2; ++j) {
            const int kk = ka + j;
            float av = 0.0f;
            if (kk < L2N) {
                const float v = l1s[m][kk];
                av = clamp01(v * v * (255.0f / 256.0f));
            } else if (kk < 2 * L2N) {
                av = clamp01(l1s[m][kk - L2N]);
            }
            a[j]  = av;
            bA[j] = (kk < 2 * L2N) ? w2[m * (2 * L2N) + kk] : 0.0f;
            bB[j] = (kk < 2 * L2N) ? w2[(m + 16) * (2 * L2N) + kk] : 0.0f;
        }
        c0 = __builtin_amdgcn_wmma_f32_16x16x4_f32(false, a, false, bA, (short)0, c0, false, false);
        c1 = __builtin_amdgcn_wmma_f32_16x16x4_f32(false, a, false, bB, (short)0, c1, false, false);
    }

    // l3 partial per lane: this lane owns output neurons n=m and n=m+16
    const float w3a = out_w[e * L3N + m];
    const float w3b = out_w[e * L3N + m + 16];
    const float b2a = l2_b[e * L3N + m];
    const float b2b = l2_b[e * L3N + m + 16];
#pragma unroll
    for (int r = 0; r < 8; ++r)
        red[r + 8 * hi][m] = clamp01(c0[r] + b2a) * w3a +
                             clamp01(c1[r] + b2b) * w3b;
    __syncthreads();

    // reduce 16 partials per token, add out_b and the raw l1 channel 15
    if (lane < 16) {
        const int row = row0 + lane;
        if (row < end) {
            float s = 0.0f;
#pragma unroll
            for (int j = 0; j < 16; ++j) s += red[lane][j];
            s += out_b[e] + l1s[lane][L2N];
            out[perm[row]] = s;
        }
    }
}

// ---------------------------------------------------------------------------
extern "C" void kernel_launch(void* const* d_in, const int* in_sizes, int n_in,
                              void* d_out, int out_size, void* d_ws, size_t ws_size,
                              hipStream_t stream) {
    const float* expert_input = (const float*)d_in[0];
    const float* router_input = (const float*)d_in[1];
    const float* router_w     = (const float*)d_in[2];
    const float* router_b     = (const float*)d_in[3];
    const float* l1_w         = (const float*)d_in[4];
    const float* l1_b         = (const float*)d_in[5];
    const float* l2_w         = (const float*)d_in[6];
    const float* l2_b         = (const float*)d_in[7];
    const float* out_w        = (const float*)d_in[8];
    const float* out_b        = (const float*)d_in[9];
    float* out = (float*)d_out;

    // workspace layout (ints)
    int* ws_i    = (int*)d_ws;
    int* counts  = ws_i;                 // 16
    int* offsets = ws_i + 16;            // 17
    int* cursor  = ws_i + 33;            // 16
    int* eidx    = ws_i + 64;            // BTOK
    int* perm    = ws_i + 64 + BTOK;     // BTOK

    zero_kernel<<<1, 64, 0, stream>>>(counts);
    router_kernel<<<BTOK / 256, 256, 0, stream>>>(router_input, router_w, router_b,
                                                  eidx, counts);
    scan_kernel<<<1, 32, 0, stream>>>(counts, offsets, cursor);
    scatter_kernel<<<BTOK / 256, 256, 0, stream>>>(eidx, cursor, perm);

    dim3 grid(TILES_PER_EXPERT, NUM_EXPERTS);
    moe_mlp_kernel<<<grid, 32, 0, stream>>>(expert_input, l1_w, l1_b, l2_w, l2_b,
                                            out_w, out_b, offsets, perm, out);
}